// GNNPolicy_77171972375190
// MI455X (gfx1250) — compile-verified
//
#include <hip/hip_runtime.h>

#define N_NODES 50000
#define N_EDGES 800000
#define NODE_DIM 256
#define HIDDEN 128
#define N_ACT 16
#define BATCH 64
#define NEG_INF -1000000.0f

typedef __attribute__((ext_vector_type(2))) float v2f;
typedef __attribute__((ext_vector_type(8))) float v8f;

// ---------------------------------------------------------------- utilities
__global__ void fill_kernel(float* __restrict__ p, float v, int n) {
    int i = blockIdx.x * blockDim.x + threadIdx.x;
    if (i < n) p[i] = v;
}

__global__ void deg_count_kernel(const int* __restrict__ dst,
                                 float* __restrict__ deg, int nE) {
    int i = blockIdx.x * blockDim.x + threadIdx.x;
    if (i < nE) atomicAdd(&deg[dst[i]], 1.0f);
}

__global__ void rsqrt_kernel(float* __restrict__ p, int n) {
    int i = blockIdx.x * blockDim.x + threadIdx.x;
    if (i < n) p[i] = rsqrtf(p[i]);
}

// ------------------------------------------------------------- fp32 WMMA GEMM
// H[rows x 128] = X[rows x K] @ W[K x 128]   (bias added later in combine)
//
// Block = 256 threads = 8 waves; block owns a 64-row stripe. Wave w owns
// output columns [16w, 16w+16) and FOUR 16x16 accumulators stacked in M, so
// each B fragment (2 global loads of W) feeds 4 V_WMMA_F32_16X16X4_F32 ops.
// K is consumed in 128-wide chunks staged in a 64x129 LDS tile (stride 129 ==
// 1 mod 64 banks -> conflict-free column reads). Exact fp32 math (matches the
// f32 reference; the roofline is dominated by the edge scatter anyway).
#define M_TILE 64
#define K_CHUNK 128

__global__ __launch_bounds__(256) void gemm_wmma_kernel(
    const float* __restrict__ X, const float* __restrict__ W,
    float* __restrict__ H, int K, int nrows) {
    __shared__ float xs[M_TILE * (K_CHUNK + 1)];   // 33 KB
    const int stride  = K_CHUNK + 1;
    const int rowBase = blockIdx.x * M_TILE;

    const int wave    = threadIdx.x >> 5;
    const int lane    = threadIdx.x & 31;
    const int half    = lane >> 4;                 // 0: K pair {0,1}; 1: {2,3}
    const int l15     = lane & 15;
    const int colBase = wave * 16;

    v8f acc0 = {}, acc1 = {}, acc2 = {}, acc3 = {};

    for (int kb = 0; kb < K; kb += K_CHUNK) {
        __syncthreads();                            // protect previous chunk
        // stage 64 x 128 chunk of X (zero-fill past nrows)
        for (int li = threadIdx.x; li < M_TILE * K_CHUNK; li += 256) {
            int r  = li >> 7;
            int c  = li & 127;
            int gr = rowBase + r;
            xs[r * stride + c] = (gr < nrows) ? X[gr * K + kb + c] : 0.0f;
        }
        // prefetch next K chunk of X while we compute on this one
        if (kb + K_CHUNK < K) {
            for (int li = threadIdx.x; li < M_TILE * K_CHUNK; li += 256 * 8) {
                int r  = li >> 7;
                int c  = li & 127;
                int gr = rowBase + r;
                if (gr < nrows)
                    __builtin_prefetch(&X[gr * K + kb + K_CHUNK + c], 0, 3);
            }
        }
        __syncthreads();

        for (int k = 0; k < K_CHUNK; k += 4) {
            const int klo = k + half * 2;
            // B 4x16: lane = col N, VGPR = K within half (reused by 4 WMMAs)
            v2f b;
            b.x = W[(kb + klo) * HIDDEN + colBase + l15];
            b.y = W[(kb + klo + 1) * HIDDEN + colBase + l15];
            // A 16x4 per M-subtile: lane = row M, VGPR = K within half
            v2f a0, a1, a2, a3;
            a0.x = xs[(0 * 16 + l15) * stride + klo];
            a0.y = xs[(0 * 16 + l15) * stride + klo + 1];
            a1.x = xs[(1 * 16 + l15) * stride + klo];
            a1.y = xs[(1 * 16 + l15) * stride + klo + 1];
            a2.x = xs[(2 * 16 + l15) * stride + klo];
            a2.y = xs[(2 * 16 + l15) * stride + klo + 1];
            a3.x = xs[(3 * 16 + l15) * stride + klo];
            a3.y = xs[(3 * 16 + l15) * stride + klo + 1];
            acc0 = __builtin_amdgcn_wmma_f32_16x16x4_f32(false, a0, false, b,
                                                         (short)0, acc0, false, false);
            acc1 = __builtin_amdgcn_wmma_f32_16x16x4_f32(false, a1, false, b,
                                                         (short)0, acc1, false, false);
            acc2 = __builtin_amdgcn_wmma_f32_16x16x4_f32(false, a2, false, b,
                                                         (short)0, acc2, false, false);
            acc3 = __builtin_amdgcn_wmma_f32_16x16x4_f32(false, a3, false, b,
                                                         (short)0, acc3, false, false);
        }
    }

    // C/D layout: VGPR i -> M = i (+8 for lanes 16..31), N = lane&15
#pragma unroll
    for (int i = 0; i < 8; ++i) {
        int rr = i + half * 8;
        int r0 = rowBase + 0 * 16 + rr;
        int r1 = rowBase + 1 * 16 + rr;
        int r2 = rowBase + 2 * 16 + rr;
        int r3 = rowBase + 3 * 16 + rr;
        if (r0 < nrows) H[r0 * HIDDEN + colBase + l15] = acc0[i];
        if (r1 < nrows) H[r1 * HIDDEN + colBase + l15] = acc1[i];
        if (r2 < nrows) H[r2 * HIDDEN + colBase + l15] = acc2[i];
        if (r3 < nrows) H[r3 * HIDDEN + colBase + l15] = acc3[i];
    }
}

// ---------------------------------------------------- edge scatter-aggregate
// One wave per edge: 32 lanes x float4 = 128 dims; h and agg are L2-resident
// (25.6 MB each << 192 MB L2), so the f32 atomics stay on-chip.
__global__ __launch_bounds__(256) void aggregate_kernel(
    const int* __restrict__ src, const int* __restrict__ dst,
    const float* __restrict__ dinv, const float* __restrict__ h,
    float* __restrict__ agg) {
    const int lane  = threadIdx.x & 31;
    const int wave  = (blockIdx.x * blockDim.x + threadIdx.x) >> 5;
    const int nWave = (gridDim.x * blockDim.x) >> 5;
    for (int e = wave; e < N_EDGES; e += nWave) {
        const int s = src[e];
        const int d = dst[e];
        const float ne = dinv[s] * dinv[d];
        const float4 v = ((const float4*)(h + s * HIDDEN))[lane];
        float* ap = agg + d * HIDDEN + lane * 4;
        atomicAdd(ap + 0, v.x * ne);
        atomicAdd(ap + 1, v.y * ne);
        atomicAdd(ap + 2, v.z * ne);
        atomicAdd(ap + 3, v.w * ne);
    }
}

// out = relu(agg + dinv^2 * h + bias)
__global__ void combine_relu_kernel(const float* __restrict__ agg,
                                    const float* __restrict__ h,
                                    const float* __restrict__ dinv,
                                    const float* __restrict__ bias,
                                    float* __restrict__ out) {
    int i = blockIdx.x * blockDim.x + threadIdx.x;
    if (i < N_NODES * HIDDEN) {
        int r = i >> 7;
        int c = i & 127;
        float di = dinv[r];
        float v = agg[i] + di * di * h[i] + bias[c];
        out[i] = fmaxf(v, 0.0f);
    }
}

// --------------------------------------------------------------- policy head
__device__ __forceinline__ float bits_to_gumbel(unsigned bits) {
    // jax.random.uniform: f = bitcast((bits>>9)|0x3f800000) - 1 in [0,1)
    float f = __uint_as_float((bits >> 9) | 0x3F800000u) - 1.0f;
    const float tiny = 1.1754943508222875e-38f;   // finfo(f32).tiny
    f = f * (1.0f - tiny) + tiny;
    f = fmaxf(f, tiny);
    return -logf(-logf(f));
}

__global__ __launch_bounds__(512) void head_kernel(
    const float* __restrict__ x, const int* __restrict__ idx,
    const int* __restrict__ mask,
    const float* __restrict__ Wa, const float* __restrict__ ba,
    const float* __restrict__ Wc, const float* __restrict__ bc,
    float* __restrict__ out) {
    __shared__ float cur[BATCH][HIDDEN];
    __shared__ float gs[BATCH * N_ACT];
    const int t = threadIdx.x;

    for (int li = t; li < BATCH * HIDDEN; li += 512) {
        int b = li >> 7;
        int k = li & 127;
        cur[b][k] = x[idx[b] * HIDDEN + k];
    }

    // threefry2x32 with key(42) = (0, 42); counts = iota(1024) split in half:
    // thread t encrypts (t, t+512) -> bits[t], bits[t+512].
    {
        const unsigned ks0 = 0u, ks1 = 42u;
        const unsigned ks2 = 0x1BD11BDAu ^ ks0 ^ ks1;
        unsigned x0 = (unsigned)t + ks0;
        unsigned x1 = (unsigned)(t + 512) + ks1;
#define TF_ROUND(r) { x0 += x1; x1 = (x1 << (r)) | (x1 >> (32 - (r))); x1 ^= x0; }
#define TF_G1 TF_ROUND(13) TF_ROUND(15) TF_ROUND(26) TF_ROUND(6)
#define TF_G2 TF_ROUND(17) TF_ROUND(29) TF_ROUND(16) TF_ROUND(24)
        TF_G1 x0 += ks1; x1 += ks2 + 1u;
        TF_G2 x0 += ks2; x1 += ks0 + 2u;
        TF_G1 x0 += ks0; x1 += ks1 + 3u;
        TF_G2 x0 += ks1; x1 += ks2 + 4u;
        TF_G1 x0 += ks2; x1 += ks0 + 5u;
#undef TF_G2
#undef TF_G1
#undef TF_ROUND
        gs[t]       = bits_to_gumbel(x0);
        gs[t + 512] = bits_to_gumbel(x1);
    }
    __syncthreads();

    if (t < BATCH) {
        const int b = t;
        float logits[N_ACT];
#pragma unroll
        for (int a = 0; a < N_ACT; ++a) {
            float s = ba[a];
            for (int k = 0; k < HIDDEN; ++k) s += cur[b][k] * Wa[k * N_ACT + a];
            logits[a] = mask[b * N_ACT + a] ? s : NEG_INF;
            out[b * N_ACT + a] = logits[a];
        }
        // categorical: argmax(logits + gumbel), first index on ties
        int act = 0;
        float best = logits[0] + gs[b * N_ACT + 0];
#pragma unroll
        for (int a = 1; a < N_ACT; ++a) {
            float v = logits[a] + gs[b * N_ACT + a];
            if (v > best) { best = v; act = a; }
        }
        // log_softmax at chosen action
        float m = logits[0];
#pragma unroll
        for (int a = 1; a < N_ACT; ++a) m = fmaxf(m, logits[a]);
        float se = 0.0f;
#pragma unroll
        for (int a = 0; a < N_ACT; ++a) se += expf(logits[a] - m);
        const float lse = m + logf(se);

        float vp = bc[0];
        for (int k = 0; k < HIDDEN; ++k) vp += cur[b][k] * Wc[k];

        out[BATCH * N_ACT + b]             = (float)act;        // act
        out[BATCH * N_ACT + BATCH + b]     = logits[act] - lse; // logp
        out[BATCH * N_ACT + 2 * BATCH + b] = vp;                // vpred
    }
}

// ------------------------------------------------------------------- driver
extern "C" void kernel_launch(void* const* d_in, const int* in_sizes, int n_in,
                              void* d_out, int out_size, void* d_ws, size_t ws_size,
                              hipStream_t stream) {
    const float* node_features = (const float*)d_in[0];
    const int*   ei            = (const int*)d_in[1];
    const int*   src           = ei;
    const int*   dst           = ei + N_EDGES;
    const int*   cur_idx       = (const int*)d_in[2];
    const int*   masks         = (const int*)d_in[3];
    const float* W1            = (const float*)d_in[4];
    const float* b1            = (const float*)d_in[5];
    const float* W2            = (const float*)d_in[6];
    const float* b2            = (const float*)d_in[7];
    const float* Wa            = (const float*)d_in[8];
    const float* ba            = (const float*)d_in[9];
    const float* Wc            = (const float*)d_in[10];
    const float* bc            = (const float*)d_in[11];
    float* o = (float*)d_out;

    float* ws   = (float*)d_ws;
    float* dinv = ws;                                 // 50000 (holds deg first)
    float* h    = ws + 50048;                         // 50000*128
    float* agg  = h + N_NODES * HIDDEN;               // 50000*128
    float* xbuf = agg + N_NODES * HIDDEN;             // 50000*128

    const int NF      = N_NODES * HIDDEN;
    const int gemmGrd = (N_NODES + M_TILE - 1) / M_TILE;

    // degree -> dinv (shared by both layers)
    fill_kernel<<<(N_NODES + 255) / 256, 256, 0, stream>>>(dinv, 1.0f, N_NODES);
    deg_count_kernel<<<(N_EDGES + 255) / 256, 256, 0, stream>>>(dst, dinv, N_EDGES);
    rsqrt_kernel<<<(N_NODES + 255) / 256, 256, 0, stream>>>(dinv, N_NODES);

    // ---- layer 1
    gemm_wmma_kernel<<<gemmGrd, 256, 0, stream>>>(node_features, W1, h, NODE_DIM, N_NODES);
    fill_kernel<<<(NF + 255) / 256, 256, 0, stream>>>(agg, 0.0f, NF);
    aggregate_kernel<<<2048, 256, 0, stream>>>(src, dst, dinv, h, agg);
    combine_relu_kernel<<<(NF + 255) / 256, 256, 0, stream>>>(agg, h, dinv, b1, xbuf);

    // ---- layer 2
    gemm_wmma_kernel<<<gemmGrd, 256, 0, stream>>>(xbuf, W2, h, HIDDEN, N_NODES);
    fill_kernel<<<(NF + 255) / 256, 256, 0, stream>>>(agg, 0.0f, NF);
    aggregate_kernel<<<2048, 256, 0, stream>>>(src, dst, dinv, h, agg);
    combine_relu_kernel<<<(NF + 255) / 256, 256, 0, stream>>>(agg, h, dinv, b2, xbuf);

    // ---- heads
    head_kernel<<<1, 512, 0, stream>>>(xbuf, cur_idx, masks, Wa, ba, Wc, bc, o);
}